// ScaleDotProductAttention_57226144252711
// MI455X (gfx1250) — compile-verified
//
#include <hip/hip_runtime.h>

typedef __attribute__((ext_vector_type(2))) float v2f;
typedef __attribute__((ext_vector_type(4))) float v4f;
typedef __attribute__((ext_vector_type(8))) float v8f;

#define B_      2
#define H_      16
#define S_      2048
#define D_      64
#define QT      16      // q rows per block
#define WAVES   8
#define THREADS 256

// One workgroup computes a 16-row Q tile for one (b,h):
//   phase 1: scores(16 x 2048) = mask ? 1e-9 : (Q K^T)/sqrt(64)   -> LDS (WMMA f32 16x16x4)
//   phase 2: row softmax over 2048, write attn to global (NT) + normalized LDS
//   phase 3: context(16 x 64) = attn @ V                          -> global (WMMA f32 16x16x4)
__global__ __launch_bounds__(THREADS)
void sdpa_wmma_kernel(const float* __restrict__ Q, const float* __restrict__ K,
                      const float* __restrict__ V, const unsigned char* __restrict__ mask,
                      float* __restrict__ ctx, float* __restrict__ attn)
{
    __shared__ float smem[QT * S_];   // 128 KB score/attn tile; reused for partial-context reduce

    const int tid  = threadIdx.x;
    const int lane = tid & 31;
    const int w    = tid >> 5;
    const int half = lane >> 4;       // which 16-lane half (K-pair select in WMMA frags)
    const int l16  = lane & 15;

    const int bh    = blockIdx.x >> 7;    // 0..31  (b*H + h)
    const int qtile = blockIdx.x & 127;   // 0..127
    const int b     = bh >> 4;
    const int qbase = qtile * QT;

    const float* Qp = Q + (size_t)bh * S_ * D_ + (size_t)qbase * D_;
    const float* Kp = K + (size_t)bh * S_ * D_;
    const float* Vp = V + (size_t)bh * S_ * D_;
    const unsigned char* Mp = mask + (size_t)b * S_ * S_;   // mask is [B,1,S,S]

    // ---- preload Q as A-fragments for V_WMMA_F32_16X16X4_F32 ----
    // A 16x4 layout: lanes 0-15 hold K=0,1 ; lanes 16-31 hold K=2,3  -> contiguous float2/lane
    v2f qa[16];
    #pragma unroll
    for (int i = 0; i < 16; ++i) {
        const float* p = Qp + l16 * D_ + 4 * i + 2 * half;
        qa[i] = v2f{p[0], p[1]};
    }

    // ---- phase 1: masked, scaled scores into LDS ----
    for (int jt = w; jt < S_ / 16; jt += WAVES) {
        const int jbase = jt * 16;
        v2f kb[16];
        #pragma unroll
        for (int i = 0; i < 16; ++i) {   // B-frag: lane holds K-col j=l16, d-pair (4i+2*half)
            const float* p = Kp + (size_t)(jbase + l16) * D_ + 4 * i + 2 * half;
            kb[i] = v2f{p[0], p[1]};
        }
        v8f acc = {};
        #pragma unroll
        for (int i = 0; i < 16; ++i) {
            acc = __builtin_amdgcn_wmma_f32_16x16x4_f32(false, qa[i], false, kb[i],
                                                        (short)0, acc, false, false);
        }
        #pragma unroll
        for (int r = 0; r < 8; ++r) {    // C/D layout: row = r + 8*half, col = l16
            const int m  = r + 8 * half;
            const int qg = qbase + m;
            const unsigned char mv = Mp[(size_t)qg * S_ + jbase + l16];
            const float s = mv ? 1e-9f : acc[r] * 0.125f;   // 1/sqrt(64)
            smem[m * S_ + jbase + l16] = s;
        }
    }
    __syncthreads();

    // ---- phase 2: softmax along S (16 threads per row), write attn ----
    {
        const int row = tid >> 4;
        const int c   = tid & 15;
        float* rowp = smem + row * S_;

        float mmax = -3.402823466e38f;
        for (int i = 0; i < S_ / 64; ++i) {
            v4f v = *reinterpret_cast<const v4f*>(rowp + i * 64 + c * 4);
            mmax = fmaxf(mmax, fmaxf(fmaxf(v.x, v.y), fmaxf(v.z, v.w)));
        }
        #pragma unroll
        for (int off = 8; off >= 1; off >>= 1)    // xor<=8 stays within 16-lane group (wave32)
            mmax = fmaxf(mmax, __shfl_xor(mmax, off, 32));

        float sum = 0.f;
        for (int i = 0; i < S_ / 64; ++i) {
            v4f v = *reinterpret_cast<v4f*>(rowp + i * 64 + c * 4);
            v.x = __expf(v.x - mmax); v.y = __expf(v.y - mmax);
            v.z = __expf(v.z - mmax); v.w = __expf(v.w - mmax);
            sum += v.x + v.y + v.z + v.w;
            *reinterpret_cast<v4f*>(rowp + i * 64 + c * 4) = v;
        }
        #pragma unroll
        for (int off = 8; off >= 1; off >>= 1)
            sum += __shfl_xor(sum, off, 32);
        const float inv = 1.0f / sum;

        // attn is 96% of output bytes and is write-once: stream it past L2 (TH=NT)
        float* aout = attn + ((size_t)bh * S_ + qbase + row) * S_;
        for (int i = 0; i < S_ / 64; ++i) {
            v4f v = *reinterpret_cast<v4f*>(rowp + i * 64 + c * 4);
            v *= inv;
            *reinterpret_cast<v4f*>(rowp + i * 64 + c * 4) = v;
            __builtin_nontemporal_store(v, reinterpret_cast<v4f*>(aout + i * 64 + c * 4));
        }
    }
    __syncthreads();

    // ---- phase 3: context = attn @ V ; each wave reduces 256 of the K dim ----
    const int kstart = w * (S_ / WAVES);
    v8f cacc[4];
    #pragma unroll
    for (int nt = 0; nt < 4; ++nt) cacc[nt] = (v8f){};

    for (int kk = 0; kk < (S_ / WAVES) / 4; ++kk) {
        const int k0 = kstart + kk * 4;
        // A-frag straight from LDS (8B-aligned float2)
        const v2f a = *reinterpret_cast<const v2f*>(&smem[l16 * S_ + k0 + 2 * half]);
        #pragma unroll
        for (int nt = 0; nt < 4; ++nt) {
            // B-frag: rows k0+2*half (+1), col = nt*16 + l16 of V
            const float* vp = Vp + (size_t)(k0 + 2 * half) * D_ + nt * 16 + l16;
            const v2f bb = v2f{vp[0], vp[D_]};
            cacc[nt] = __builtin_amdgcn_wmma_f32_16x16x4_f32(false, a, false, bb,
                                                             (short)0, cacc[nt], false, false);
        }
    }
    __syncthreads();   // all waves finished reading attn from LDS; safe to reuse

    // per-wave partials: part[w][m][d] in reused LDS (8*16*64 floats = 32 KB)
    float* part = smem;
    #pragma unroll
    for (int nt = 0; nt < 4; ++nt)
        #pragma unroll
        for (int r = 0; r < 8; ++r)
            part[w * (QT * D_) + (r + 8 * half) * D_ + nt * 16 + l16] = cacc[nt][r];
    __syncthreads();

    // reduce across the 8 waves and write context (also write-once -> NT)
    for (int it = 0; it < (QT * D_) / THREADS; ++it) {
        const int e = tid + it * THREADS;
        float s = 0.f;
        #pragma unroll
        for (int ww = 0; ww < WAVES; ++ww) s += part[ww * (QT * D_) + e];
        const int row = e >> 6, col = e & 63;
        __builtin_nontemporal_store(s, &ctx[((size_t)bh * S_ + qbase + row) * D_ + col]);
    }
}

extern "C" void kernel_launch(void* const* d_in, const int* in_sizes, int n_in,
                              void* d_out, int out_size, void* d_ws, size_t ws_size,
                              hipStream_t stream) {
    const float* Q = (const float*)d_in[0];
    const float* K = (const float*)d_in[1];
    const float* V = (const float*)d_in[2];
    const unsigned char* mask = (const unsigned char*)d_in[3];   // jnp.bool_ -> 1 byte/elem

    float* ctx  = (float*)d_out;                                 // [B,H,S,D]
    float* attn = (float*)d_out + (size_t)B_ * H_ * S_ * D_;     // [B,H,S,S]

    dim3 grid(B_ * H_ * (S_ / QT));   // 4096 workgroups
    sdpa_wmma_kernel<<<grid, THREADS, 0, stream>>>(Q, K, V, mask, ctx, attn);
}